// GaussianModel_63161789055251
// MI455X (gfx1250) — compile-verified
//
#include <hip/hip_runtime.h>
#include <hip/hip_bf16.h>
#include <math.h>

typedef float v2f __attribute__((ext_vector_type(2)));
typedef float v8f __attribute__((ext_vector_type(8)));

#define RES     256
#define TILE    16
#define NT      16          // tiles per dimension (RES/TILE)
#define NG      1024        // gaussians
#define CHUNK   16
#define NCHUNK  (NG / CHUNK)
#define LOG2E   1.4426950408889634f

__device__ __forceinline__ float fast_exp2(float x) {
#if __has_builtin(__builtin_amdgcn_exp2f)
    return __builtin_amdgcn_exp2f(x);   // v_exp_f32 (base-2)
#else
    return exp2f(x);
#endif
}

__device__ __forceinline__ float fast_log2(float x) {
#if __has_builtin(__builtin_amdgcn_logf)
    return __builtin_amdgcn_logf(x);    // v_log_f32 (base-2)
#else
    return log2f(x);
#endif
}

__device__ __forceinline__ v2f mk2(float a, float b) { v2f v; v.x = a; v.y = b; return v; }

// ---------------------------------------------------------------------------
// Kernel 1: per-gaussian preprocessing.
// Produces, per gaussian:
//   coef[8]: log2-scaled bilinear coefficients of power over features
//            [px^2, py^2, px*py, px, py, 1, 0, 0], with log2(opacity) folded
//            into the constant term  =>  w*op = exp2(P . G)
//   bnds[4]: clipped bounding box (rminx, rminy, rmaxx, rmaxy) for tile cull
// ---------------------------------------------------------------------------
__global__ void gs_preprocess(const float* __restrict__ xyz,
                              const float* __restrict__ scaling,
                              const float* __restrict__ rotation,
                              const float* __restrict__ opacity,
                              const float* __restrict__ rot,
                              float* __restrict__ coef,   // NG*8
                              float* __restrict__ bnds)   // NG*4
{
    int n = blockIdx.x * blockDim.x + threadIdx.x;
    if (n >= NG) return;

    // quaternion -> rotation matrix
    float qr = rotation[n*4+0], qx = rotation[n*4+1];
    float qy = rotation[n*4+2], qz = rotation[n*4+3];
    float inv = rsqrtf(qr*qr + qx*qx + qy*qy + qz*qz);
    qr *= inv; qx *= inv; qy *= inv; qz *= inv;

    float R[9];
    R[0] = 1.f - 2.f*(qy*qy + qz*qz); R[1] = 2.f*(qx*qy - qr*qz); R[2] = 2.f*(qx*qz + qr*qy);
    R[3] = 2.f*(qx*qy + qr*qz); R[4] = 1.f - 2.f*(qx*qx + qz*qz); R[5] = 2.f*(qy*qz - qr*qx);
    R[6] = 2.f*(qx*qz - qr*qy); R[7] = 2.f*(qy*qz + qr*qx); R[8] = 1.f - 2.f*(qx*qx + qy*qy);

    float s0 = scaling[n*3+0], s1 = scaling[n*3+1], s2 = scaling[n*3+2];
    float L[9];
    L[0] = R[0]*s0; L[1] = R[1]*s1; L[2] = R[2]*s2;
    L[3] = R[3]*s0; L[4] = R[4]*s1; L[5] = R[5]*s2;
    L[6] = R[6]*s0; L[7] = R[7]*s1; L[8] = R[8]*s2;

    // cov3d = L * L^T (symmetric)
    float C[9];
    #pragma unroll
    for (int i = 0; i < 3; ++i)
        #pragma unroll
        for (int j = 0; j < 3; ++j)
            C[3*i + j] = L[3*i+0]*L[3*j+0] + L[3*i+1]*L[3*j+1] + L[3*i+2]*L[3*j+2];

    // A = diag(res,res,0) @ rot ; keep rows 0,1
    float A0[3], A1[3];
    #pragma unroll
    for (int j = 0; j < 3; ++j) { A0[j] = (float)RES * rot[j]; A1[j] = (float)RES * rot[3+j]; }

    // cov2d = A C A^T (2x2)
    float v0[3], v1[3];
    #pragma unroll
    for (int i = 0; i < 3; ++i) {
        v0[i] = C[3*i+0]*A0[0] + C[3*i+1]*A0[1] + C[3*i+2]*A0[2];
        v1[i] = C[3*i+0]*A1[0] + C[3*i+1]*A1[1] + C[3*i+2]*A1[2];
    }
    float c00 = A0[0]*v0[0] + A0[1]*v0[1] + A0[2]*v0[2];
    float c10 = A1[0]*v0[0] + A1[1]*v0[1] + A1[2]*v0[2];
    float c01 = A0[0]*v1[0] + A0[1]*v1[1] + A0[2]*v1[2];
    float c11 = A1[0]*v1[0] + A1[1]*v1[1] + A1[2]*v1[2];

    float det = c00*c11 - c01*c10;
    float mid = 0.5f*(c00 + c11);
    float sq  = sqrtf(fmaxf(mid*mid - det, 0.1f));
    float radii = ceilf(3.0f * sqrtf(fmaxf(mid + sq, mid - sq)));

    float mx = xyz[n*3+0] * (RES * 0.5f);
    float my = xyz[n*3+1] * (RES * 0.5f);

    bnds[n*4+0] = fminf(fmaxf(mx - radii, 0.f), (float)RES - 1.f);
    bnds[n*4+1] = fminf(fmaxf(my - radii, 0.f), (float)RES - 1.f);
    bnds[n*4+2] = fminf(fmaxf(mx + radii, 0.f), (float)RES - 1.f);
    bnds[n*4+3] = fminf(fmaxf(my + radii, 0.f), (float)RES - 1.f);

    float idet = 1.0f / det;
    float a  = c11 * idet;
    float b  = c00 * idet;
    float cc = -(c01 + c10) * idet;    // i01 + i10

    float op = opacity[n];
    coef[n*8+0] = -0.5f * LOG2E * a;
    coef[n*8+1] = -0.5f * LOG2E * b;
    coef[n*8+2] = -0.5f * LOG2E * cc;
    coef[n*8+3] = LOG2E * (a*mx + 0.5f*cc*my);
    coef[n*8+4] = LOG2E * (b*my + 0.5f*cc*mx);
    coef[n*8+5] = LOG2E * (-0.5f*(mx*mx*a + my*my*b + mx*my*cc)) + fast_log2(op);
    coef[n*8+6] = 0.f;
    coef[n*8+7] = 0.f;
}

// ---------------------------------------------------------------------------
// Kernel 2: one 16x16 pixel tile per block; 8 waves, 2 pixel rows per wave.
// power(16 px x 16 gaussians) via two chained V_WMMA_F32_16X16X4_F32 (K=8),
// then w = exp2(power) * tile_mask(lane) accumulated into the D-layout regs.
// ---------------------------------------------------------------------------
__global__ __launch_bounds__(256) void gs_splat(const float* __restrict__ coef,
                                                const float* __restrict__ bnds,
                                                float* __restrict__ out)
{
    __shared__ float  sC[NG * 8];          // 32 KB coefficient table
    __shared__ float4 sB[NG];              // 16 KB bounding boxes
    __shared__ float  sRed[8][2][256];     // 16 KB per-wave reduction scratch

    const int tid = threadIdx.x;

    // Stage whole table into LDS (every block touches every gaussian)
    {
        const float4* c4  = (const float4*)coef;
        float4*       sc4 = (float4*)sC;
        for (int i = tid; i < NG * 2; i += 256) sc4[i] = c4[i];
        const float4* b4 = (const float4*)bnds;
        for (int i = tid; i < NG; i += 256) sB[i] = b4[i];
    }
    __syncthreads();

    const int tile = blockIdx.x;
    const int tx = tile & (NT - 1);
    const int ty = tile >> 4;
    const int w    = tid >> 5;
    const int lane = tid & 31;
    const int col  = lane & 15;       // N index (gaussian within chunk)
    const bool lo  = (lane < 16);

    const float tsx = (float)(tx * TILE);
    const float tsy = (float)(ty * TILE);

    // pixel coordinates: coords[i] = i - 128 + 0.5
    const float px  = tsx + (float)col - 127.5f;
    const int ry0 = 2*w, ry1 = 2*w + 1;
    const float py0 = tsy + (float)ry0 - 127.5f;
    const float py1 = tsy + (float)ry1 - 127.5f;

    // A matrices (16x4 f32 layout: lanes 0-15 hold K={0,1}, lanes 16-31 K={2,3})
    // features: K = [px^2, py^2, px*py, px, py, 1, 0, 0]
    v2f A0r0, A1r0, A0r1, A1r1;
    if (lo) {
        A0r0 = mk2(px*px,  py0*py0); A1r0 = mk2(py0, 1.0f);
        A0r1 = mk2(px*px,  py1*py1); A1r1 = mk2(py1, 1.0f);
    } else {
        A0r0 = mk2(px*py0, px);      A1r0 = mk2(0.f, 0.f);
        A0r1 = mk2(px*py1, px);      A1r1 = mk2(0.f, 0.f);
    }

    v8f acc0 = {0.f,0.f,0.f,0.f,0.f,0.f,0.f,0.f};
    v8f acc1 = {0.f,0.f,0.f,0.f,0.f,0.f,0.f,0.f};

    for (int c = 0; c < NCHUNK; ++c) {
        const int g = c * CHUNK + col;

        // tile-vs-bbox cull (strict >, matching reference)
        const float4 bb = sB[g];
        const bool ok = (fminf(bb.z, tsx + 15.f) > fmaxf(bb.x, tsx)) &&
                        (fminf(bb.w, tsy + 15.f) > fmaxf(bb.y, tsy));
#if __has_builtin(__builtin_amdgcn_ballot_w32)
        const unsigned int bal = __builtin_amdgcn_ballot_w32(ok);
#else
        const unsigned int bal = (unsigned int)__ballot(ok);
#endif
        if (bal == 0u) continue;              // uniform branch: EXEC stays all-ones

        const float maskv = ok ? 1.0f : 0.0f; // per-lane == per-N-column mask

        // B matrix (4x16 f32 layout: lanes 0-15 hold K={0,1}, lanes 16-31 K={2,3})
        const float4* g4 = (const float4*)(sC + g * 8);
        const float4 cA = g4[0];   // g0..g3
        const float4 cB = g4[1];   // g4,g5,0,0
        const v2f B0 = lo ? mk2(cA.x, cA.y) : mk2(cA.z, cA.w);
        const v2f B1 = lo ? mk2(cB.x, cB.y) : mk2(0.f, 0.f);

        const v8f z = {0.f,0.f,0.f,0.f,0.f,0.f,0.f,0.f};
        // power = A(K=4..7) x B(K=4..7) + A(K=0..3) x B(K=0..3)
        v8f p0 = __builtin_amdgcn_wmma_f32_16x16x4_f32(false, A1r0, false, B1, (short)0, z,  false, false);
        p0     = __builtin_amdgcn_wmma_f32_16x16x4_f32(false, A0r0, false, B0, (short)0, p0, false, false);
        v8f p1 = __builtin_amdgcn_wmma_f32_16x16x4_f32(false, A1r1, false, B1, (short)0, z,  false, false);
        p1     = __builtin_amdgcn_wmma_f32_16x16x4_f32(false, A0r1, false, B0, (short)0, p1, false, false);

        #pragma unroll
        for (int i = 0; i < 8; ++i) {
            acc0[i] += maskv * fast_exp2(p0[i]);
            acc1[i] += maskv * fast_exp2(p1[i]);
        }
    }

    // Reduce over N (gaussian columns): D layout puts N in lanes, so bounce
    // through LDS and sum the 16 lanes of each half.
    #pragma unroll
    for (int i = 0; i < 8; ++i) {
        sRed[w][0][lane * 8 + i] = acc0[i];
        sRed[w][1][lane * 8 + i] = acc1[i];
    }
    __syncthreads();

    if (lane < 16) {
        const int x = lane;                        // pixel x within tile
        const int r = (x < 8) ? x : (x - 8);       // acc VGPR index
        const int sbase = (x < 8) ? 0 : 16;        // lane half holding this M
        float s0v = 0.f, s1v = 0.f;
        #pragma unroll
        for (int s = 0; s < 16; ++s) {
            s0v += sRed[w][0][(sbase + s) * 8 + r];
            s1v += sRed[w][1][(sbase + s) * 8 + r];
        }
        const int px_idx = tx * TILE + x;
        out[(ty * TILE + ry0) * RES + px_idx] = s0v;
        out[(ty * TILE + ry1) * RES + px_idx] = s1v;
    }
}

// ---------------------------------------------------------------------------
extern "C" void kernel_launch(void* const* d_in, const int* in_sizes, int n_in,
                              void* d_out, int out_size, void* d_ws, size_t ws_size,
                              hipStream_t stream) {
    const float* xyz      = (const float*)d_in[0];
    const float* scaling  = (const float*)d_in[1];
    const float* rotation = (const float*)d_in[2];
    const float* opacity  = (const float*)d_in[3];
    const float* rot      = (const float*)d_in[4];
    // d_in[5] = res (known 256)

    float* coef = (float*)d_ws;          // NG*8 floats
    float* bnds = coef + NG * 8;         // NG*4 floats

    gs_preprocess<<<(NG + 255) / 256, 256, 0, stream>>>(
        xyz, scaling, rotation, opacity, rot, coef, bnds);

    gs_splat<<<NT * NT, 256, 0, stream>>>(coef, bnds, (float*)d_out);
}